// LinearMemAttentionWithScoreOutput_56014963475095
// MI455X (gfx1250) — compile-verified
//
#include <hip/hip_runtime.h>
#include <math.h>

// Problem constants (B=1)
#define SEQ   2048
#define DM    1024
#define NH    16
#define HD    64

#define PSTRIDE 2052   // 32-row LDS score stripe, 4-bank row skew (2052 % 64 == 4)
#define TSTRIDE 68     // K/Q/V tile row stride (68 % 64 == 4 -> skewed, conflict-free)

// LDS layout (floats): P[32*2052] | Qs[32*68] (phase2: red) | KB0[64*68] | KB1[64*68]
#define SMEM_FLOATS (32 * PSTRIDE + 32 * TSTRIDE + 2 * 64 * TSTRIDE)
#define SMEM_BYTES  (SMEM_FLOATS * 4)

typedef __attribute__((ext_vector_type(2))) float v2f;
typedef __attribute__((ext_vector_type(4))) float v4f;
typedef __attribute__((ext_vector_type(8))) float v8f;

// D = A(16x4 f32) * B(4x16 f32) + C(16x16 f32), exact fp32 matrix op on gfx1250.
__device__ __forceinline__ v8f wmma_f32(v2f a, v2f b, v8f c) {
  return __builtin_amdgcn_wmma_f32_16x16x4_f32(false, a, false, b, (short)0, c,
                                               false, false);
}

// CDNA5 async copy: memory -> LDS without VGPR round trip (tracked by ASYNCcnt).
// GVS mode: mem_addr = SADDR(u64) + VADDR(i32); VDST holds per-lane LDS byte addr.
__device__ __forceinline__ void async_b128(unsigned lds_off, unsigned gvoff,
                                           const float* base) {
  asm volatile("global_load_async_to_lds_b128 %0, %1, %2"
               :: "v"(lds_off), "v"(gvoff), "s"(base)
               : "memory");
}
// Generic pointers into LDS carry the addrspace(3) offset in their low 32 bits.
__device__ __forceinline__ unsigned ldsaddr(const float* p) {
  return (unsigned)(uintptr_t)p;
}
#define WAIT_ASYNC(NHEX) asm volatile("s_wait_asynccnt " NHEX ::: "memory")

// ---------------------------------------------------------------------------
// Kernel 1: Y = X @ W for W in {Wq, Wk, Wv} selected by blockIdx.z.
// 8 wave32 per block, 32(M) x 64(N) tile, one 16x16 WMMA tile per wave,
// register-double-buffered K staging.
// ---------------------------------------------------------------------------
__global__ __launch_bounds__(256) void qkv_gemm(
    const float* __restrict__ X,
    const float* __restrict__ Wq, const float* __restrict__ Wk,
    const float* __restrict__ Wv,
    float* __restrict__ Q, float* __restrict__ Kb, float* __restrict__ Vb) {
  __shared__ float Xs[32][36];  // [m][k]
  __shared__ float Ws[32][68];  // [k][n]

  const float* W;
  float* Y;
  if (blockIdx.z == 0)      { W = Wq; Y = Q;  }
  else if (blockIdx.z == 1) { W = Wk; Y = Kb; }
  else                      { W = Wv; Y = Vb; }

  const int m0 = blockIdx.y * 32;
  const int n0 = blockIdx.x * 64;
  const int tid = threadIdx.x;
  const int lane = tid & 31;
  const int wave = tid >> 5;
  const int mw = (wave >> 2) * 16;
  const int nw = (wave & 3) * 16;
  const int lg = lane >> 4;
  const int l15 = lane & 15;

  const int xr = tid >> 3, xc = (tid & 7) * 4;
  const int wr = tid >> 4, wc = (tid & 15) * 4;

  v8f acc = {};
  float4 xreg, wreg0, wreg1;

  xreg  = *reinterpret_cast<const float4*>(X + (size_t)(m0 + xr) * DM + xc);
  wreg0 = *reinterpret_cast<const float4*>(W + (size_t)(wr) * DM + n0 + wc);
  wreg1 = *reinterpret_cast<const float4*>(W + (size_t)(wr + 16) * DM + n0 + wc);

  for (int k0 = 0; k0 < DM; k0 += 32) {
    __syncthreads();
    Xs[xr][xc + 0] = xreg.x; Xs[xr][xc + 1] = xreg.y;
    Xs[xr][xc + 2] = xreg.z; Xs[xr][xc + 3] = xreg.w;
    Ws[wr][wc + 0] = wreg0.x; Ws[wr][wc + 1] = wreg0.y;
    Ws[wr][wc + 2] = wreg0.z; Ws[wr][wc + 3] = wreg0.w;
    Ws[wr + 16][wc + 0] = wreg1.x; Ws[wr + 16][wc + 1] = wreg1.y;
    Ws[wr + 16][wc + 2] = wreg1.z; Ws[wr + 16][wc + 3] = wreg1.w;
    __syncthreads();

    if (k0 + 32 < DM) {
      xreg  = *reinterpret_cast<const float4*>(X + (size_t)(m0 + xr) * DM + k0 + 32 + xc);
      wreg0 = *reinterpret_cast<const float4*>(W + (size_t)(k0 + 32 + wr) * DM + n0 + wc);
      wreg1 = *reinterpret_cast<const float4*>(W + (size_t)(k0 + 48 + wr) * DM + n0 + wc);
    }

#pragma unroll
    for (int kk = 0; kk < 32; kk += 4) {
      v2f a, b;
      a.x = Xs[mw + l15][kk + 2 * lg + 0];
      a.y = Xs[mw + l15][kk + 2 * lg + 1];
      b.x = Ws[kk + 2 * lg + 0][nw + l15];
      b.y = Ws[kk + 2 * lg + 1][nw + l15];
      acc = wmma_f32(a, b, acc);
    }
  }

  const int mrow = m0 + mw + lg * 8;
  const int ncol = n0 + nw + l15;
#pragma unroll
  for (int j = 0; j < 8; ++j)
    Y[(size_t)(mrow + j) * DM + ncol] = acc[j];
}

// ---------------------------------------------------------------------------
// Kernel 2: fused attention for one (head, 32-q-row stripe).
//   phase 1: S = (Q_h.K_h^T)/8 -> LDS stripe P[32][2052]; K chunks stream in
//            via GLOBAL_LOAD_ASYNC_TO_LDS_B128, ping-ponged across 2 buffers.
//   phase 2: row softmax in LDS; probs leave via non-temporal b128 stores
//            (keeps the 256 MB stream out of L2 so K/V re-reads stay cached).
//   phase 3: context = P @ V_h; V chunks async-staged, first two issued
//            before softmax so the fetch overlaps it.
// ---------------------------------------------------------------------------
__global__ __launch_bounds__(256) void fused_attn(
    const float* __restrict__ Q, const float* __restrict__ Kb,
    const float* __restrict__ Vb, float* __restrict__ Pr,
    float* __restrict__ Ctx) {
  extern __shared__ float smem[];
  float* P   = smem;                        // [32][PSTRIDE]
  float* Qs  = smem + 32 * PSTRIDE;         // [32][68]  (phase 2: red scratch)
  float* KB0 = Qs + 32 * TSTRIDE;           // [64][68]  K/V ping
  float* KB1 = KB0 + 64 * TSTRIDE;          // [64][68]  K/V pong
  float* red = Qs;

  const int h  = blockIdx.x;
  const int q0 = blockIdx.y * 32;
  const int tid = threadIdx.x;
  const int lane = tid & 31;
  const int wave = tid >> 5;
  const int mw = (wave >> 2) * 16;  // q sub-tile
  const int nw = (wave & 3) * 16;   // k / d sub-tile
  const int lg = lane >> 4;
  const int l15 = lane & 15;
  const size_t pbase = (size_t)h * SEQ * SEQ;

  const int sr = tid >> 4, sc = (tid & 15) * 4;  // 16-row x 64-col stager

  const float* baseK = Kb + h * HD;
  const float* baseV = Vb + h * HD;

  // ---- phase 1: scores into LDS stripe --------------------------------
  // issue K chunks 0 and 1 (row-major [k][d], skewed stride => no transpose)
#pragma unroll
  for (int p = 0; p < 4; ++p) {
    int r = sr + p * 16;
    async_b128(ldsaddr(KB0 + r * TSTRIDE + sc),
               (unsigned)(((0 * 64 + r) * DM + sc) * sizeof(float)), baseK);
  }
#pragma unroll
  for (int p = 0; p < 4; ++p) {
    int r = sr + p * 16;
    async_b128(ldsaddr(KB1 + r * TSTRIDE + sc),
               (unsigned)(((1 * 64 + r) * DM + sc) * sizeof(float)), baseK);
  }

  // stage Q rows (32 x 64) once
#pragma unroll
  for (int p = 0; p < 2; ++p) {
    float4 x = *reinterpret_cast<const float4*>(
        Q + (size_t)(q0 + sr + p * 16) * DM + h * HD + sc);
    Qs[(sr + p * 16) * TSTRIDE + sc + 0] = x.x;
    Qs[(sr + p * 16) * TSTRIDE + sc + 1] = x.y;
    Qs[(sr + p * 16) * TSTRIDE + sc + 2] = x.z;
    Qs[(sr + p * 16) * TSTRIDE + sc + 3] = x.w;
  }

  for (int kc = 0; kc < SEQ / 64; ++kc) {
    if (kc + 1 < SEQ / 64) { WAIT_ASYNC("0x4"); } else { WAIT_ASYNC("0x0"); }
    __syncthreads();  // chunk kc visible to all waves; Qs staged (first iter)

    const float* kbuf = (kc & 1) ? KB1 : KB0;
    v8f acc = {};
#pragma unroll
    for (int d = 0; d < HD; d += 4) {
      v2f a, b;
      a.x = Qs[(mw + l15) * TSTRIDE + d + 2 * lg + 0];
      a.y = Qs[(mw + l15) * TSTRIDE + d + 2 * lg + 1];
      // B[d][k] = K[k][d] read straight from row-major tile (skewed stride)
      b.x = kbuf[(nw + l15) * TSTRIDE + d + 2 * lg + 0];
      b.y = kbuf[(nw + l15) * TSTRIDE + d + 2 * lg + 1];
      acc = wmma_f32(a, b, acc);
    }
    const int kcol = kc * 64 + nw + l15;
#pragma unroll
    for (int j = 0; j < 8; ++j)
      P[(mw + lg * 8 + j) * PSTRIDE + kcol] = acc[j] * 0.125f;

    __syncthreads();  // all waves done reading KB[kc&1]
    if (kc + 2 < SEQ / 64) {
      float* nbuf = (kc & 1) ? KB1 : KB0;  // KB[(kc+2)&1] == KB[kc&1]
#pragma unroll
      for (int p = 0; p < 4; ++p) {
        int r = sr + p * 16;
        async_b128(ldsaddr(nbuf + r * TSTRIDE + sc),
                   (unsigned)((((kc + 2) * 64 + r) * DM + sc) * sizeof(float)),
                   baseK);
      }
    }
  }

  // prefetch V chunks 0 and 1 via async engine; they fly during softmax
#pragma unroll
  for (int p = 0; p < 2; ++p) {
    int r = sr + p * 16;
    async_b128(ldsaddr(KB0 + r * TSTRIDE + sc),
               (unsigned)(((0 * 32 + r) * DM + sc) * sizeof(float)), baseV);
  }
#pragma unroll
  for (int p = 0; p < 2; ++p) {
    int r = sr + p * 16;
    async_b128(ldsaddr(KB1 + r * TSTRIDE + sc),
               (unsigned)(((1 * 32 + r) * DM + sc) * sizeof(float)), baseV);
  }

  // ---- phase 2: softmax in LDS + non-temporal probs writeout ----------
  {
    const int row = tid >> 3;  // 8 threads per row
    const int t   = tid & 7;
    float* prow = P + row * PSTRIDE;

    float m = -INFINITY;
#pragma unroll 8
    for (int j = 0; j < SEQ / 8; ++j) m = fmaxf(m, prow[t + 8 * j]);
    red[tid] = m;
    __syncthreads();
    if (tid < 32) {
      float mm = red[tid * 8];
#pragma unroll
      for (int i = 1; i < 8; ++i) mm = fmaxf(mm, red[tid * 8 + i]);
      red[256 + tid] = mm;
    }
    __syncthreads();
    m = red[256 + row];

    float sum = 0.f;
#pragma unroll 8
    for (int j = 0; j < SEQ / 8; ++j) {
      float e = __expf(prow[t + 8 * j] - m);
      prow[t + 8 * j] = e;
      sum += e;
    }
    red[tid] = sum;
    __syncthreads();
    if (tid < 32) {
      float s = red[tid * 8];
#pragma unroll
      for (int i = 1; i < 8; ++i) s += red[tid * 8 + i];
      red[256 + tid] = s;
    }
    __syncthreads();
    const float inv = 1.0f / red[256 + row];
#pragma unroll 8
    for (int j = 0; j < SEQ / 8; ++j) prow[t + 8 * j] *= inv;
  }
  __syncthreads();

  // coalesced streaming writeout: b128 + TH_STORE_NT (don't pollute L2)
#pragma unroll 4
  for (int i = 0; i < 64; ++i) {
    int flat = tid + i * 256;  // 16384 float4s = 32 x 2048 floats
    int r  = flat >> 9;
    int c4 = (flat & 511) * 4;
    v4f v = *reinterpret_cast<v4f*>(P + r * PSTRIDE + c4);
    __builtin_nontemporal_store(
        v, reinterpret_cast<v4f*>(Pr + pbase + (size_t)(q0 + r) * SEQ + c4));
  }

  // ---- phase 3: context = P @ V_h, P read from LDS --------------------
  v8f acc = {};
  for (int kb = 0; kb < SEQ / 32; ++kb) {
    if (kb + 1 < SEQ / 32) { WAIT_ASYNC("0x2"); } else { WAIT_ASYNC("0x0"); }
    __syncthreads();  // V chunk kb visible; probs writeout reads of P done

    const float* vbuf = (kb & 1) ? KB1 : KB0;
    const int k0 = kb * 32;
#pragma unroll
    for (int kk = 0; kk < 32; kk += 4) {
      v2f a, b;
      a.x = P[(mw + l15) * PSTRIDE + k0 + kk + 2 * lg + 0];
      a.y = P[(mw + l15) * PSTRIDE + k0 + kk + 2 * lg + 1];
      b.x = vbuf[(kk + 2 * lg + 0) * TSTRIDE + nw + l15];
      b.y = vbuf[(kk + 2 * lg + 1) * TSTRIDE + nw + l15];
      acc = wmma_f32(a, b, acc);
    }
    __syncthreads();  // all waves done reading KB[kb&1]

    if (kb + 2 < SEQ / 32) {
      float* nbuf = (kb & 1) ? KB1 : KB0;  // KB[(kb+2)&1]
#pragma unroll
      for (int p = 0; p < 2; ++p) {
        int r = sr + p * 16;
        async_b128(ldsaddr(nbuf + r * TSTRIDE + sc),
                   (unsigned)((((kb + 2) * 32 + r) * DM + sc) * sizeof(float)),
                   baseV);
      }
    }
  }

  const int qrow = q0 + mw + lg * 8;
  const int dcol = h * HD + nw + l15;
#pragma unroll
  for (int j = 0; j < 8; ++j)
    Ctx[(size_t)(qrow + j) * DM + dcol] = acc[j];
}

// ---------------------------------------------------------------------------
extern "C" void kernel_launch(void* const* d_in, const int* in_sizes, int n_in,
                              void* d_out, int out_size, void* d_ws, size_t ws_size,
                              hipStream_t stream) {
  const float* X  = (const float*)d_in[0];
  const float* Wq = (const float*)d_in[1];
  const float* Wk = (const float*)d_in[2];
  const float* Wv = (const float*)d_in[3];

  float* out = (float*)d_out;
  float* Ctx = out;                          // [2048][1024] context
  float* Pr  = out + (size_t)SEQ * DM;       // [16][2048][2048] probs

  float* ws = (float*)d_ws;                  // 24 MB: Q,K,V fp32
  float* Q  = ws;
  float* Kb = ws + (size_t)SEQ * DM;
  float* Vb = ws + 2 * (size_t)SEQ * DM;

  // ~299 KB of dynamic LDS (CDNA5 WGP allows up to 320 KB per workgroup)
  static int lds_cfg = 0;
  if (!lds_cfg) {
    hipFuncSetAttribute(reinterpret_cast<const void*>(fused_attn),
                        hipFuncAttributeMaxDynamicSharedMemorySize, SMEM_BYTES);
    lds_cfg = 1;
  }

  dim3 blk(256);
  qkv_gemm<<<dim3(DM / 64, SEQ / 32, 3), blk, 0, stream>>>(X, Wq, Wk, Wv, Q, Kb, Vb);
  fused_attn<<<dim3(NH, SEQ / 32), blk, SMEM_BYTES, stream>>>(Q, Kb, Vb, Pr, Ctx);
}